// Block_71820443124300
// MI455X (gfx1250) — compile-verified
//
#include <hip/hip_runtime.h>
#include <math.h>

// ---------------------------------------------------------------------------
// MI455X (gfx1250) implementation. fp32 throughout, matrix ops via
// V_WMMA_F32_16X16X4_F32 (wave32, 16x16 C tile = 8 VGPRs/lane).
// B operands staged transposed in LDS so A and B fragments are both single
// contiguous 8-byte DS loads (no cross-row repacking movs before each WMMA).
// ---------------------------------------------------------------------------

typedef __attribute__((ext_vector_type(2))) float v2f;
typedef __attribute__((ext_vector_type(8))) float v8f;

#define WMMA_F32(a, b, c) \
    __builtin_amdgcn_wmma_f32_16x16x4_f32(false, (a), false, (b), (short)0, (c), false, false)

enum { MODE_BIAS = 0, MODE_RESADD = 1, MODE_GELU = 2, MODE_MOE = 3 };

__device__ __forceinline__ float gelu_tanh(float x) {
    float x3 = x * x * x;
    return 0.5f * x * (1.0f + tanhf(0.7978845608028654f * (x + 0.044715f * x3)));
}

// ---------------------------------------------------------------------------
// Generic tiled GEMM:  C[M,N] = epilogue(A[M,K] @ W[K,N] (+bias))
// Block 256 threads = 8 waves (wave32). Tile BM=64, BN=128, BK=32.
// Wave (wm in 0..3, wn in 0..1) owns rows [wm*16,+16) x cols [wn*64,+64)
// = 4 C tiles of 16x16 (4 x v8f accumulators).
// ---------------------------------------------------------------------------
#define BM 64
#define BN 128
#define BK 32
#define APAD 36   // 36*4B rows: 16 lanes hit 16 distinct banks, 8B/16B aligned
#define BPAD 36   // same property for the transposed B tile

__global__ __launch_bounds__(256) void k_gemm(
    const float* __restrict__ A, const float* __restrict__ W,
    const float* __restrict__ bias, const float* __restrict__ src,
    float* __restrict__ C,
    int M, int N, int K, int ldw,
    int mode, const float* __restrict__ gates, int expert, int S_tok)
{
    __shared__ float As[BM][APAD];     // As[m][k]
    __shared__ float Bts[BN][BPAD];    // Bts[n][k]  (transposed W tile)

    const int tid  = threadIdx.x;
    const int wave = tid >> 5;
    const int lane = tid & 31;
    const int half = lane >> 4;
    const int l16  = lane & 15;
    const int wm   = wave & 3;      // 0..3  -> M sub-tile
    const int wn   = wave >> 2;     // 0..1  -> N sub-tile
    const int mb   = blockIdx.y * BM;
    const int nb   = blockIdx.x * BN;

    v8f acc[4];
#pragma unroll
    for (int t = 0; t < 4; ++t)
#pragma unroll
        for (int v = 0; v < 8; ++v) acc[t][v] = 0.0f;

    for (int kb = 0; kb < K; kb += BK) {
        // A tile: 64x32 floats = 512 float4 slots, 2 per thread
#pragma unroll
        for (int i = 0; i < 2; ++i) {
            int slot = tid + i * 256;
            int r = slot >> 3, c4 = slot & 7;
            float4 val = *(const float4*)(A + (size_t)(mb + r) * K + kb + c4 * 4);
            *(float4*)&As[r][c4 * 4] = val;
        }
        // W tile: 32x128 floats, stored transposed Bts[n][k]
#pragma unroll
        for (int i = 0; i < 4; ++i) {
            int slot = tid + i * 256;
            int r = slot >> 5, c4 = slot & 31;
            float4 val = *(const float4*)(W + (size_t)(kb + r) * ldw + nb + c4 * 4);
            Bts[c4 * 4 + 0][r] = val.x;
            Bts[c4 * 4 + 1][r] = val.y;
            Bts[c4 * 4 + 2][r] = val.z;
            Bts[c4 * 4 + 3][r] = val.w;
        }
        // Prefetch next K-tile into cache while this one is consumed.
        if (kb + BK < K) {
#pragma unroll
            for (int i = 0; i < 2; ++i) {
                int slot = tid + i * 256;
                int r = slot >> 3, c4 = slot & 7;
                if (c4 == 0)
                    __builtin_prefetch(A + (size_t)(mb + r) * K + kb + BK, 0, 1);
            }
#pragma unroll
            for (int i = 0; i < 4; ++i) {
                int slot = tid + i * 256;
                int r = slot >> 5, c4 = slot & 31;
                if ((c4 & 7) == 0)
                    __builtin_prefetch(W + (size_t)(kb + BK + r) * ldw + nb + c4 * 4, 0, 1);
            }
        }
        __syncthreads();

#pragma unroll
        for (int k0 = 0; k0 < BK; k0 += 4) {
            const v2f a = *(const v2f*)&As[wm * 16 + l16][k0 + 2 * half];
#pragma unroll
            for (int t = 0; t < 4; ++t) {
                const v2f b = *(const v2f*)&Bts[wn * 64 + t * 16 + l16][k0 + 2 * half];
                acc[t] = WMMA_F32(a, b, acc[t]);
            }
        }
        __syncthreads();
    }

    // Epilogue. 16-row tiles never straddle a batch boundary (S_tok % 64 == 0).
    float gsc = 1.0f;
    if (mode == MODE_MOE) {
        int row0 = mb + wm * 16;
        gsc = gates[(row0 / S_tok) * 8 + expert];
    }
#pragma unroll
    for (int t = 0; t < 4; ++t) {
        int col = nb + wn * 64 + t * 16 + l16;
        float bv = bias ? bias[col] : 0.0f;
#pragma unroll
        for (int v = 0; v < 8; ++v) {
            int row = mb + wm * 16 + v + 8 * half;
            size_t idx = (size_t)row * N + col;
            float val = acc[t][v] + bv;
            if (mode == MODE_BIAS)        C[idx] = val;
            else if (mode == MODE_RESADD) C[idx] = val + src[idx];
            else if (mode == MODE_GELU)   C[idx] = gelu_tanh(val);
            else                          C[idx] += gsc * val;   // sequential, no race
        }
    }
}

// ---------------------------------------------------------------------------
// Flash attention, no mask. One block = 128 threads = 4 waves; each wave owns
// 16 query rows (QT=64 per block). K/V streamed in KT=32 chunks through LDS.
// Q pre-scaled by 1/sqrt(DH)=0.125. V kept transposed (Vts[n][k]) so the PV
// B-fragment is a contiguous v2f load; K kept row-major (d contiguous) so the
// QK^T B-fragment is also contiguous.
// ---------------------------------------------------------------------------
#define ADH  64
#define QT   64
#define KT   32
#define DPAD 68   // 68*4B rows -> 16 lanes, distinct banks; float4-aligned
#define VPAD 36
#define PPAD 36

__device__ __forceinline__ float rmax16(float x) {
#pragma unroll
    for (int m = 1; m < 16; m <<= 1) x = fmaxf(x, __shfl_xor(x, m, 16));
    return x;
}
__device__ __forceinline__ float rsum16(float x) {
#pragma unroll
    for (int m = 1; m < 16; m <<= 1) x += __shfl_xor(x, m, 16);
    return x;
}

__global__ __launch_bounds__(128) void k_attn(
    const float* __restrict__ qkv, float* __restrict__ o,
    int Bn, int Sn, int Dn, int Hn)
{
    __shared__ float Qs[QT][DPAD];     // Qs[q][d]
    __shared__ float Ks[KT][DPAD];     // Ks[k][d]
    __shared__ float Vts[ADH][VPAD];   // Vts[d_out][k]  (transposed V tile)
    __shared__ float Ps[4][16][PPAD];  // per-wave P scratch (16 x KT)

    const int b  = blockIdx.x / Hn;
    const int h  = blockIdx.x % Hn;
    const int qb = blockIdx.y * QT;
    const int tid  = threadIdx.x;
    const int wave = tid >> 5;
    const int lane = tid & 31;
    const int half = lane >> 4;
    const int l16  = lane & 15;

    const size_t ld = (size_t)3 * Dn;                       // qkv row stride
    const float* qbase = qkv + (size_t)b * Sn * ld + (size_t)h * ADH;
    const float* kbase = qbase + Dn;
    const float* vbase = qbase + 2 * Dn;

    // Q tile: 64x64 floats = 1024 float4 slots, 8 per thread, scaled by 1/8
    for (int i = tid; i < QT * (ADH / 4); i += 128) {
        int r = i >> 4, c4 = i & 15;
        float4 q4 = *(const float4*)(qbase + (size_t)(qb + r) * ld + c4 * 4);
        q4.x *= 0.125f; q4.y *= 0.125f; q4.z *= 0.125f; q4.w *= 0.125f;
        *(float4*)&Qs[r][c4 * 4] = q4;
    }

    v8f oacc[4];
    float mrow[8], lrow[8];
#pragma unroll
    for (int t = 0; t < 4; ++t)
#pragma unroll
        for (int v = 0; v < 8; ++v) oacc[t][v] = 0.0f;
#pragma unroll
    for (int v = 0; v < 8; ++v) { mrow[v] = -1e30f; lrow[v] = 0.0f; }

    __syncthreads();

    for (int kb = 0; kb < Sn; kb += KT) {
        // K tile (row-major) + V tile (transposed): 512 float4 slots each
        for (int i = tid; i < KT * (ADH / 4); i += 128) {
            int r = i >> 4, c4 = i & 15;
            float4 k4 = *(const float4*)(kbase + (size_t)(kb + r) * ld + c4 * 4);
            *(float4*)&Ks[r][c4 * 4] = k4;
            float4 v4 = *(const float4*)(vbase + (size_t)(kb + r) * ld + c4 * 4);
            Vts[c4 * 4 + 0][r] = v4.x;
            Vts[c4 * 4 + 1][r] = v4.y;
            Vts[c4 * 4 + 2][r] = v4.z;
            Vts[c4 * 4 + 3][r] = v4.w;
        }
        // Prefetch next K/V tile (one line per row).
        if (kb + KT < Sn) {
            for (int i = tid; i < KT; i += 128) {
                __builtin_prefetch(kbase + (size_t)(kb + KT + i) * ld, 0, 1);
                __builtin_prefetch(vbase + (size_t)(kb + KT + i) * ld, 0, 1);
            }
        }
        __syncthreads();

        // S = Q @ K^T  (16 q-rows x KT cols, 2 tiles of 16x16, K-dim = 64)
        v8f s[2];
#pragma unroll
        for (int t = 0; t < 2; ++t) {
            v8f a2;
#pragma unroll
            for (int v = 0; v < 8; ++v) a2[v] = 0.0f;
#pragma unroll
            for (int d0 = 0; d0 < ADH; d0 += 4) {
                const v2f a  = *(const v2f*)&Qs[wave * 16 + l16][d0 + 2 * half];
                const v2f bb = *(const v2f*)&Ks[t * 16 + l16][d0 + 2 * half];
                a2 = WMMA_F32(a, bb, a2);
            }
            s[t] = a2;
        }

        // online softmax over the KT columns
#pragma unroll
        for (int v = 0; v < 8; ++v) {
            float mn = fmaxf(s[0][v], s[1][v]);
            mn = rmax16(mn);
            float mc = fmaxf(mrow[v], mn);
            float alpha = __expf(mrow[v] - mc);
            float p0 = __expf(s[0][v] - mc);
            float p1 = __expf(s[1][v] - mc);
            Ps[wave][v + 8 * half][0 * 16 + l16] = p0;
            Ps[wave][v + 8 * half][1 * 16 + l16] = p1;
            float ps = rsum16(p0 + p1);
            lrow[v] = lrow[v] * alpha + ps;
            mrow[v] = mc;
#pragma unroll
            for (int t = 0; t < 4; ++t) oacc[t][v] *= alpha;
        }
        // Wave-private LDS RAW: per-wave DS ops are in-order; no block sync needed.

        // O += P @ V  (16 x 64, K-dim = KT)
#pragma unroll
        for (int t2 = 0; t2 < 4; ++t2) {
#pragma unroll
            for (int p0 = 0; p0 < KT; p0 += 4) {
                const v2f a  = *(const v2f*)&Ps[wave][l16][p0 + 2 * half];
                const v2f bb = *(const v2f*)&Vts[t2 * 16 + l16][p0 + 2 * half];
                oacc[t2] = WMMA_F32(a, bb, oacc[t2]);
            }
        }
        __syncthreads();   // before K/V tiles are overwritten
    }

#pragma unroll
    for (int t = 0; t < 4; ++t)
#pragma unroll
        for (int v = 0; v < 8; ++v) {
            int row = qb + wave * 16 + v + 8 * half;
            o[((size_t)b * Sn + row) * Dn + h * ADH + t * 16 + l16] = oacc[t][v] / lrow[v];
        }
}

// ---------------------------------------------------------------------------
// Router: logits[b,e] = mean_s( x[b,s,:] . Wr[:,e] ) + br[e]
// ---------------------------------------------------------------------------
__global__ __launch_bounds__(128) void k_router(
    const float* __restrict__ x, const float* __restrict__ Wr,
    const float* __restrict__ br, float* __restrict__ logits,
    int Bn, int Sn, int Dn, int En)
{
    const int e = blockIdx.x % En;
    const int b = blockIdx.x / En;
    __shared__ float red[128];
    float acc = 0.0f;
    for (int s = threadIdx.x; s < Sn; s += 128) {
        const float* xr = x + ((size_t)b * Sn + s) * Dn;
        float dot = 0.0f;
        for (int d = 0; d < Dn; ++d) dot += xr[d] * Wr[d * En + e];
        acc += dot;
    }
    red[threadIdx.x] = acc;
    __syncthreads();
    for (int st = 64; st > 0; st >>= 1) {
        if (threadIdx.x < st) red[threadIdx.x] += red[threadIdx.x + st];
        __syncthreads();
    }
    if (threadIdx.x == 0) logits[b * En + e] = red[0] / (float)Sn + br[e];
}

__global__ void k_gates(const float* __restrict__ logits, float* __restrict__ gates,
                        int Bn, int En)
{
    int b = threadIdx.x;
    if (b >= Bn) return;
    float mx = -1e30f;
    for (int e = 0; e < En; ++e) mx = fmaxf(mx, logits[b * En + e]);
    float ex[16], s = 0.0f;
    for (int e = 0; e < En; ++e) { ex[e] = __expf(logits[b * En + e] - mx); s += ex[e]; }
    for (int e = 0; e < En; ++e) gates[b * En + e] = ex[e] / s;
}

// ---------------------------------------------------------------------------
// Global LayerNorm: scalar mean / unbiased var over the whole tensor.
// ---------------------------------------------------------------------------
__global__ __launch_bounds__(256) void k_redstats(
    const float* __restrict__ a, const float* __restrict__ b,
    float* __restrict__ stats, size_t n)
{
    float s = 0.0f, ss = 0.0f;
    for (size_t i = (size_t)blockIdx.x * 256 + threadIdx.x; i < n;
         i += (size_t)gridDim.x * 256) {
        float t = a[i] + (b ? b[i] : 0.0f);
        s += t; ss += t * t;
    }
    __shared__ float r1[256], r2[256];
    r1[threadIdx.x] = s; r2[threadIdx.x] = ss;
    __syncthreads();
    for (int st = 128; st > 0; st >>= 1) {
        if (threadIdx.x < st) { r1[threadIdx.x] += r1[threadIdx.x + st];
                                r2[threadIdx.x] += r2[threadIdx.x + st]; }
        __syncthreads();
    }
    if (threadIdx.x == 0) {
        atomicAdd(&stats[0], r1[0]);
        atomicAdd(&stats[1], r2[0]);
    }
}

__global__ __launch_bounds__(256) void k_lnapply(
    const float* __restrict__ a, const float* __restrict__ b,
    const float* __restrict__ stats, const float* __restrict__ w,
    const float* __restrict__ bias, float* __restrict__ out,
    size_t n, int Dn)
{
    float mean = stats[0] / (float)n;
    float var  = (stats[1] - stats[0] * stats[0] / (float)n) / (float)(n - 1);
    float rstd = rsqrtf(var + 1e-12f);
    for (size_t i = (size_t)blockIdx.x * 256 + threadIdx.x; i < n;
         i += (size_t)gridDim.x * 256) {
        float t = a[i] + (b ? b[i] : 0.0f);
        int d = (int)(i % (size_t)Dn);
        out[i] = (t - mean) * rstd * w[d] + bias[d];
    }
}

__global__ void k_zero(float* __restrict__ p, size_t n)
{
    for (size_t i = (size_t)blockIdx.x * blockDim.x + threadIdx.x; i < n;
         i += (size_t)gridDim.x * blockDim.x) p[i] = 0.0f;
}

// ---------------------------------------------------------------------------
extern "C" void kernel_launch(void* const* d_in, const int* in_sizes, int n_in,
                              void* d_out, int out_size, void* d_ws, size_t ws_size,
                              hipStream_t stream)
{
    (void)in_sizes; (void)n_in; (void)out_size; (void)ws_size;
    const int B = 2, S = 2048, D = 512, H = 8, E = 8, FH = 4 * D;
    const int M = B * S;                 // 4096 tokens
    const size_t nBSD = (size_t)M * D;   // 2,097,152

    const float* x     = (const float*)d_in[0];
    const float* W1    = (const float*)d_in[1];
    const float* b1    = (const float*)d_in[2];
    const float* W2    = (const float*)d_in[3];
    const float* b2    = (const float*)d_in[4];
    const float* Wr    = (const float*)d_in[5];
    const float* br    = (const float*)d_in[6];
    const float* Wfc   = (const float*)d_in[7];
    const float* bfc   = (const float*)d_in[8];
    const float* Wproj = (const float*)d_in[9];
    const float* bproj = (const float*)d_in[10];
    const float* ln1w  = (const float*)d_in[11];
    const float* ln1b  = (const float*)d_in[12];
    const float* ln2w  = (const float*)d_in[13];
    const float* ln2b  = (const float*)d_in[14];

    // Workspace arena (floats). MoE h/m buffers reuse the dead QKV region.
    float* ws     = (float*)d_ws;
    float* stats  = ws;               // [0..1]=LN1, [2..3]=LN2
    float* logits = ws + 4;           // 16
    float* gates  = ws + 20;          // 16
    float* qkv    = ws + 64;          // M*3D
    float* hbuf   = qkv;              // M*512   (reuse)
    float* mbuf   = qkv + nBSD;       // M*D     (reuse)
    float* obuf   = qkv + (size_t)M * 3 * D;
    float* abuf   = obuf + nBSD;
    float* xln1   = abuf + nBSD;

    k_zero<<<1, 64, 0, stream>>>(stats, 64);

    // 1) qkv = x @ W1 + b1                          [4096,1536]
    k_gemm<<<dim3(3 * D / BN, M / BM), 256, 0, stream>>>(
        x, W1, b1, nullptr, qkv, M, 3 * D, D, 3 * D, MODE_BIAS, nullptr, 0, S);

    // 2) attention -> obuf                          [4096,512]
    k_attn<<<dim3(B * H, S / QT), 128, 0, stream>>>(qkv, obuf, B, S, D, H);

    // 3) abuf = obuf @ W2 + b2 + x   (residual)
    k_gemm<<<dim3(D / BN, M / BM), 256, 0, stream>>>(
        obuf, W2, b2, x, abuf, M, D, D, D, MODE_RESADD, nullptr, 0, S);

    // 4) global LayerNorm 1 -> xln1
    k_redstats<<<512, 256, 0, stream>>>(abuf, nullptr, stats + 0, nBSD);
    k_lnapply<<<512, 256, 0, stream>>>(abuf, nullptr, stats + 0, ln1w, ln1b, xln1, nBSD, D);

    // 5) router + gates
    k_router<<<B * E, 128, 0, stream>>>(xln1, Wr, br, logits, B, S, D, E);
    k_gates<<<1, 32, 0, stream>>>(logits, gates, B, E);

    // 6) dense MoE, F-chunked (FC=512), gate-weighted accumulate into mbuf
    k_zero<<<1024, 256, 0, stream>>>(mbuf, nBSD);
    const int FC = 512;
    for (int e = 0; e < E; ++e) {
        const float* wfc_e = Wfc   + (size_t)e * D * FH;
        const float* wpj_e = Wproj + (size_t)e * FH * D;
        for (int c = 0; c < FH / FC; ++c) {
            // h = gelu(xln1 @ Wfc[e][:, c*FC:(c+1)*FC] + bfc[e][chunk])
            k_gemm<<<dim3(FC / BN, M / BM), 256, 0, stream>>>(
                xln1, wfc_e + c * FC, bfc + (size_t)e * FH + c * FC, nullptr,
                hbuf, M, FC, D, FH, MODE_GELU, nullptr, 0, S);
            // m += gates[b,e] * (h @ Wproj[e][chunk,:] (+ bproj[e] on chunk 0))
            k_gemm<<<dim3(D / BN, M / BM), 256, 0, stream>>>(
                hbuf, wpj_e + (size_t)c * FC * D,
                (c == 0) ? (bproj + (size_t)e * D) : nullptr, nullptr,
                mbuf, M, D, FC, D, MODE_MOE, gates, e, S);
        }
    }

    // 7) global LayerNorm 2 over (mbuf + xln1) -> d_out
    k_redstats<<<512, 256, 0, stream>>>(mbuf, xln1, stats + 2, nBSD);
    k_lnapply<<<512, 256, 0, stream>>>(mbuf, xln1, stats + 2, ln2w, ln2b,
                                       (float*)d_out, nBSD, D);
}